// WarpController_64201171141045
// MI455X (gfx1250) — compile-verified
//
#include <hip/hip_runtime.h>
#include <hip/hip_bf16.h>

#define NL   478            // landmarks
#define NP   481            // NL + 3 (affine)
#define NC   483            // NP + 2 rhs columns
#define STRIDE 512          // padded row stride in ws (floats)
#define HH   512
#define WW   512
#define NB   2
#define REGC 1e-6f
#define EPSC 1e-10f
#define HALF_LN2 0.34657359028f   // 0.5 * ln(2): phi = r2 * HALF_LN2 * log2(r2)

typedef float v2f __attribute__((ext_vector_type(2)));
typedef float v8f __attribute__((ext_vector_type(8)));
typedef unsigned int v4u __attribute__((ext_vector_type(4)));
typedef int v4i __attribute__((ext_vector_type(4)));
typedef int v8i __attribute__((ext_vector_type(8)));

// ---------------------------------------------------------------------------
// Build augmented TPS system: [A+reg*I  K | rhs ; K^T 0 | 0]  (481 x 483)
// ---------------------------------------------------------------------------
__global__ void tps_build(const float* __restrict__ src,
                          const float* __restrict__ dst,
                          float* __restrict__ ws) {
  const int b = blockIdx.y;
  const float* s = src + (size_t)b * NL * 2;
  const float* d = dst + (size_t)b * NL * 2;
  float* Mg = ws + (size_t)b * NP * STRIDE;

  int idx = blockIdx.x * blockDim.x + threadIdx.x;
  const int total = NP * NC;
  if (idx >= total) return;
  const int r = idx / NC, c = idx % NC;

  float val = 0.0f;
  if (r < NL && c < NL) {
    const float ry = d[2*r] * 512.0f, rx = d[2*r+1] * 512.0f;
    const float cy = d[2*c] * 512.0f, cx = d[2*c+1] * 512.0f;
    const float dy = ry - cy, dx = rx - cx;
    const float r2 = fmaxf(dy*dy + dx*dx, 0.0f);
    val = (r2 * HALF_LN2) * __builtin_amdgcn_logf(fmaxf(r2, EPSC));
    if (r == c) val += REGC;
  } else if (r < NL && c >= NL && c < NP) {
    const int j = c - NL;
    val = (j == 0) ? d[2*r] * 512.0f : (j == 1) ? d[2*r+1] * 512.0f : 1.0f;
  } else if (r >= NL && c < NL) {
    const int j = r - NL;
    val = (j == 0) ? d[2*c] * 512.0f : (j == 1) ? d[2*c+1] * 512.0f : 1.0f;
  } else if (c >= NP) {
    const int ch = c - NP;
    val = (r < NL) ? (d[2*r + ch] - s[2*r + ch]) * 512.0f : 0.0f;
  }
  Mg[(size_t)r * STRIDE + c] = val;
}

// ---------------------------------------------------------------------------
// Gauss-Jordan with partial pivoting; one 1024-thread workgroup per batch.
// Solution ends up in columns NP..NP+1.
// ---------------------------------------------------------------------------
__global__ void __launch_bounds__(1024) tps_solve(float* __restrict__ ws) {
  const int b = blockIdx.x;
  float* Mg = ws + (size_t)b * NP * STRIDE;
  __shared__ float s_piv[NC];
  __shared__ float s_rv[1024];
  __shared__ int   s_ri[1024];
  const int tid = threadIdx.x;

  for (int p = 0; p < NP; ++p) {
    // --- partial pivot search over column p, rows >= p ---
    float best = -1.0f; int bi = p;
    for (int r = p + tid; r < NP; r += 1024) {
      const float v = fabsf(Mg[(size_t)r * STRIDE + p]);
      if (v > best) { best = v; bi = r; }
    }
    s_rv[tid] = best; s_ri[tid] = bi;
    __syncthreads();
    for (int off = 512; off > 0; off >>= 1) {
      if (tid < off && s_rv[tid + off] > s_rv[tid]) {
        s_rv[tid] = s_rv[tid + off]; s_ri[tid] = s_ri[tid + off];
      }
      __syncthreads();
    }
    const int prow = s_ri[0];
    const float pv = Mg[(size_t)prow * STRIDE + p];
    const float inv = 1.0f / pv;
    __syncthreads();

    // --- swap rows p <-> prow, normalize pivot row, cache in LDS ---
    for (int c = p + tid; c < NC; c += 1024) {
      const float vp = Mg[(size_t)p    * STRIDE + c];
      const float vq = Mg[(size_t)prow * STRIDE + c];
      const float nv = vq * inv;
      Mg[(size_t)prow * STRIDE + c] = vp;   // write order matters when prow==p:
      Mg[(size_t)p    * STRIDE + c] = nv;   // second store wins -> normalized row
      s_piv[c] = nv;
    }
    __syncthreads();

    // --- eliminate column p from every other row ---
    for (int r = tid; r < NP; r += 1024) {
      if (r == p) continue;
      const float f = Mg[(size_t)r * STRIDE + p];
      if (f != 0.0f) {
        for (int c = p; c < NC; ++c)
          Mg[(size_t)r * STRIDE + c] -= f * s_piv[c];
      }
    }
    __syncthreads();
  }
}

// ---------------------------------------------------------------------------
// Flow evaluation + bilinear warp. Each wave handles 32 query points as two
// M=16 tiles; phi values feed V_WMMA_F32_16X16X4_F32 accumulating rbf @ w.
//  A (16x4 f32): lanes 0-15 hold K=kb,kb+1 ; lanes 16-31 hold K=kb+2,kb+3
//  B (4x16 f32): row K = kb+2*half+component, col N = lane&15 (N=0,1 used)
//  C (16x16 f32): VGPR r -> (M = r + 8*half, N = lane&15)
// Spline weights are staged LDS-side by the Tensor Data Mover (2-wide x 478
// row tile, row stride 512 floats, straight out of the solver workspace).
// ---------------------------------------------------------------------------
__global__ void __launch_bounds__(256) tps_warp(const float* __restrict__ img,
                                                const float* __restrict__ dstlm,
                                                const float* __restrict__ ws,
                                                float* __restrict__ out) {
  const int b = blockIdx.y;
  const float* image = img + (size_t)b * HH * WW * 3;
  float* outb        = out + (size_t)b * HH * WW * 3;
  const float* lmt   = dstlm + (size_t)b * NL * 2;
  const float* sol   = ws + (size_t)b * NP * STRIDE;

  __shared__ float s_lm[960];    // scaled (y,x) landmarks, zero padded to 480
  __shared__ float s_w[1024];    // spline weights w (478x2) via TDM, zero pad
  __shared__ float s_v[8];       // affine v (3x2)
  __shared__ float s_flow[512];  // per-block flow (256 queries x 2)

  const int q = blockIdx.x * 256 + threadIdx.x;         // this thread's query
  __builtin_prefetch(image + (size_t)q * 3, 0, 3);      // global_prefetch_b8

  for (int i = threadIdx.x; i < 960; i += 256)
    s_lm[i] = (i < NL * 2) ? lmt[i] * 512.0f : 0.0f;
  for (int i = NL * 2 + threadIdx.x; i < 1024; i += 256)
    s_w[i] = 0.0f;                                      // zero the tail pad
  if (threadIdx.x < 6)
    s_v[threadIdx.x] =
        sol[(size_t)(NL + (threadIdx.x >> 1)) * STRIDE + NP + (threadIdx.x & 1)];

  // --- TDM: async-copy the 478x2 weight tile (row stride 512 f32) into LDS ---
  if (threadIdx.x == 0) {
    const unsigned long long ga = (unsigned long long)(uintptr_t)(sol + NP);
    const unsigned int lds = (unsigned int)(uintptr_t)(void*)s_w; // flat[31:0]=LDS
    v4u g0;
    g0.x = 1u;                                           // count=1, no gather
    g0.y = lds;                                          // lds_addr
    g0.z = (unsigned int)ga;                             // global_addr[31:0]
    g0.w = ((unsigned int)(ga >> 32) & 0x01FFFFFFu) | (2u << 30); // addr|type=2
    v8i g1;
    g1[0] = 0x00020000;          // wg_mask=0, data_size=4B
    g1[1] = (int)(2u << 16);     // tensor_dim0 = 2
    g1[2] = (int)(478u << 16);   // tensor_dim1 = 478 (lo16)
    g1[3] = (int)(2u << 16);     // tile_dim0 = 2
    g1[4] = 478;                 // tile_dim1 = 478, tile_dim2 = 0
    g1[5] = 512;                 // tensor_dim0_stride = 512 (lo32)
    g1[6] = 0;                   // stride hi, tensor_dim1_stride lo
    g1[7] = 0;
    const v4i gz = {0, 0, 0, 0};
#if defined(__clang_major__) && __clang_major__ >= 23
    const v8i gz8 = {0, 0, 0, 0, 0, 0, 0, 0};
    __builtin_amdgcn_tensor_load_to_lds(g0, g1, gz, gz, gz8, 0);
#else
    __builtin_amdgcn_tensor_load_to_lds(g0, g1, gz, gz, 0);
#endif
    __builtin_amdgcn_s_wait_tensorcnt(0);
  }
  __syncthreads();

  const int lane = threadIdx.x & 31;
  const int wave = threadIdx.x >> 5;
  const int hf   = lane >> 4;        // which half of the wave
  const int n    = lane & 15;        // N column / M row within tile
  const int qbase = (blockIdx.x * 8 + wave) * 32;

  const int m0 = qbase + n;          // tile0 query handled by this lane's A row
  const int m1 = m0 + 16;            // tile1
  const float q0y = (float)(m0 >> 9), q0x = (float)(m0 & 511);
  const float q1y = (float)(m1 >> 9), q1x = (float)(m1 & 511);
  const float nmask = (n < 2) ? 1.0f : 0.0f;  // one v_cndmask, hoisted

  v8f c0 = {0.f, 0.f, 0.f, 0.f, 0.f, 0.f, 0.f, 0.f};
  v8f c1 = c0;

#pragma unroll 2
  for (int kb = 0; kb < 480; kb += 4) {
    const int k0 = kb + 2 * hf;
    const float l0y = s_lm[2*k0],     l0x = s_lm[2*k0 + 1];
    const float l1y = s_lm[2*k0 + 2], l1x = s_lm[2*k0 + 3];

    v2f a0, a1, bb;
    float dy = q0y - l0y, dx = q0x - l0x, r2 = dy*dy + dx*dx;
    a0.x = (r2 * HALF_LN2) * __builtin_amdgcn_logf(fmaxf(r2, EPSC));
    dy = q0y - l1y; dx = q0x - l1x; r2 = dy*dy + dx*dx;
    a0.y = (r2 * HALF_LN2) * __builtin_amdgcn_logf(fmaxf(r2, EPSC));
    dy = q1y - l0y; dx = q1x - l0x; r2 = dy*dy + dx*dx;
    a1.x = (r2 * HALF_LN2) * __builtin_amdgcn_logf(fmaxf(r2, EPSC));
    dy = q1y - l1y; dx = q1x - l1x; r2 = dy*dy + dx*dx;
    a1.y = (r2 * HALF_LN2) * __builtin_amdgcn_logf(fmaxf(r2, EPSC));

    // Unconditional LDS loads (pad is zero); mask by mul, no EXEC traffic.
    // Rows k>=478 hold zero weights, so the tail contributes exactly zero.
    bb.x = nmask * s_w[2*k0 + n];
    bb.y = nmask * s_w[2*k0 + 2 + n];

    c0 = __builtin_amdgcn_wmma_f32_16x16x4_f32(false, a0, false, bb,
                                               (short)0, c0, false, false);
    c1 = __builtin_amdgcn_wmma_f32_16x16x4_f32(false, a1, false, bb,
                                               (short)0, c1, false, false);
  }

  // Scatter accumulators: only N=0,1 columns carry flow_y / flow_x.
  if (n < 2) {
#pragma unroll
    for (int r = 0; r < 8; ++r) {
      const int ml = r + 8 * hf;                       // 0..15 within tile
      s_flow[wave * 64 + 2 * ml + n]        = c0[r];   // tile0 queries
      s_flow[wave * 64 + 2 * (16 + ml) + n] = c1[r];   // tile1 queries
    }
  }
  __syncthreads();

  // One query per thread: affine term + bilinear sample at grid - flow.
  const float qy = (float)(q >> 9), qx = (float)(q & 511);
  const float fy = s_flow[2 * threadIdx.x]     + qy * s_v[0] + qx * s_v[2] + s_v[4];
  const float fx = s_flow[2 * threadIdx.x + 1] + qy * s_v[1] + qx * s_v[3] + s_v[5];
  const float sy = qy - fy, sx = qx - fx;

  float y0f = fminf(fmaxf(floorf(sy), 0.0f), (float)(HH - 2));
  float x0f = fminf(fmaxf(floorf(sx), 0.0f), (float)(WW - 2));
  const int y0 = (int)y0f, x0 = (int)x0f;
  const float ay = fminf(fmaxf(sy - y0f, 0.0f), 1.0f);
  const float ax = fminf(fmaxf(sx - x0f, 0.0f), 1.0f);

  const size_t base = ((size_t)y0 * WW + x0) * 3;
#pragma unroll
  for (int ch = 0; ch < 3; ++ch) {
    const float tl = image[base + ch];
    const float tr = image[base + 3 + ch];
    const float bl = image[base + (size_t)WW * 3 + ch];
    const float br = image[base + (size_t)WW * 3 + 3 + ch];
    const float top = tl + ax * (tr - tl);
    const float bot = bl + ax * (br - bl);
    outb[(size_t)q * 3 + ch] = top + ay * (bot - top);
  }
}

// ---------------------------------------------------------------------------
extern "C" void kernel_launch(void* const* d_in, const int* in_sizes, int n_in,
                              void* d_out, int out_size, void* d_ws, size_t ws_size,
                              hipStream_t stream) {
  const float* img = (const float*)d_in[0];   // images_rendered (2,512,512,3)
  const float* src = (const float*)d_in[1];   // landmarks_source (2,478,2)
  const float* dst = (const float*)d_in[2];   // landmarks_target (2,478,2)
  float* out = (float*)d_out;
  float* ws  = (float*)d_ws;                  // needs 2*481*512*4 ~= 1.9 MB

  dim3 gb((NP * NC + 255) / 256, NB);
  tps_build<<<gb, 256, 0, stream>>>(src, dst, ws);
  tps_solve<<<dim3(NB), 1024, 0, stream>>>(ws);
  tps_warp<<<dim3(HH * WW / 256, NB), 256, 0, stream>>>(img, dst, ws, out);
}